// EdgeDegreeEmbeddingNetwork_86749749444867
// MI455X (gfx1250) — compile-verified
//
#include <hip/hip_runtime.h>
#include <hip/hip_bf16.h>
#include <math.h>

typedef float v2f __attribute__((ext_vector_type(2)));
typedef float v8f __attribute__((ext_vector_type(8)));

static __device__ __forceinline__ v8f wmma4(v2f a, v2f b, v8f c) {
  // V_WMMA_F32_16X16X4_F32 : D = A(16x4 f32) x B(4x16 f32) + C(16x16 f32)
  return __builtin_amdgcn_wmma_f32_16x16x4_f32(false, a, false, b, (short)0, c,
                                               false, false);
}

// ---------------- kernel 0: zero the output (harness poisons d_out) --------
__global__ void zero_out_kernel(float4* __restrict__ out, int n4) {
  int i = blockIdx.x * blockDim.x + threadIdx.x;
  if (i < n4) out[i] = make_float4(0.f, 0.f, 0.f, 0.f);
}

// ---------------- kernel 1: fold weights ----------------------------------
// sv[c] = exp_w[c] + exp_b[c]
// Qt[d][k]  (224 x 64): d<128  : sum_c rad_w2[k][c]     * sv[c] * proj_w0[c][d]
//                       d<192  : sum_c rad_w2[k][128+c] * sv[c] * proj_w1[c][d-128]
//                       else   : sum_c rad_w2[k][256+c] * sv[c] * proj_w2[c][d-192]
// qb[d]     (224)     : same with rad_b2 replacing rad_w2 row
// W1t[n][k] (64 x 64) : rad_w1[k][n]   (transposed for B-fragment reads)
__global__ void fold_weights_kernel(
    const float* __restrict__ exp_w, const float* __restrict__ exp_b,
    const float* __restrict__ rad_w1, const float* __restrict__ rad_w2,
    const float* __restrict__ rad_b2,
    const float* __restrict__ proj_w0, const float* __restrict__ proj_w1,
    const float* __restrict__ proj_w2,
    float* __restrict__ Qt, float* __restrict__ qb, float* __restrict__ W1t) {
  const int b = blockIdx.x;
  const int t = threadIdx.x;  // 64 threads
  if (b < 224) {
    const int d = b, k = t;
    const float* wrow = rad_w2 + (size_t)k * 384;
    float s = 0.f;
    if (d < 128) {
      for (int c = 0; c < 128; ++c)
        s = fmaf(wrow[c] * (exp_w[c] + exp_b[c]), proj_w0[c * 128 + d], s);
    } else if (d < 192) {
      const int dd = d - 128;
      for (int c = 0; c < 128; ++c)
        s = fmaf(wrow[128 + c] * (exp_w[c] + exp_b[c]), proj_w1[c * 64 + dd], s);
    } else {
      const int dd = d - 192;
      for (int c = 0; c < 128; ++c)
        s = fmaf(wrow[256 + c] * (exp_w[c] + exp_b[c]), proj_w2[c * 32 + dd], s);
    }
    Qt[d * 64 + k] = s;
  } else if (b < 228) {
    const int d = (b - 224) * 64 + t;
    if (d < 224) {
      float s = 0.f;
      if (d < 128) {
        for (int c = 0; c < 128; ++c)
          s = fmaf(rad_b2[c] * (exp_w[c] + exp_b[c]), proj_w0[c * 128 + d], s);
      } else if (d < 192) {
        const int dd = d - 128;
        for (int c = 0; c < 128; ++c)
          s = fmaf(rad_b2[128 + c] * (exp_w[c] + exp_b[c]), proj_w1[c * 64 + dd], s);
      } else {
        const int dd = d - 192;
        for (int c = 0; c < 128; ++c)
          s = fmaf(rad_b2[256 + c] * (exp_w[c] + exp_b[c]), proj_w2[c * 32 + dd], s);
      }
      qb[d] = s;
    }
  } else {
    const int n = b - 228, k = t;
    W1t[n * 64 + k] = rad_w1[k * 64 + n];
  }
}

// ---------------- kernel 2: fused per-edge network + scatter ---------------
// One wave32 per 16-edge tile.
//   GEMM1: X(16x64) @ W1(64x64)  -> h   (wmma f32 16x16x4, 4 Ntiles x 16 K)
//   LayerNorm(64) + SiLU         (cross-lane xor-shuffle reductions)
//   GEMM2: h(16x64) @ Q(64x224)  -> u   (14 Ntiles x 16 K)
//   scatter: 480 features per edge via global_atomic_add_f32
#define W1_PITCH 66
#define Q_PITCH 66
#define H_PITCH 68

__global__ __launch_bounds__(256) void edge_kernel(
    const float* __restrict__ edge_attr,     // (E,9)
    const float* __restrict__ edge_scalars,  // (E,64)
    const int* __restrict__ edge_dst,        // (E)
    const float* __restrict__ rad_b1,        // (64)
    const float* __restrict__ gamma,         // (64)
    const float* __restrict__ beta,          // (64)
    const float* __restrict__ proj_b0,       // (128)
    const float* __restrict__ Qt,            // (224,64) folded
    const float* __restrict__ qb,            // (224)
    const float* __restrict__ W1t,           // (64,64) transposed
    float* __restrict__ out,                 // (N,480)
    int n_edges) {
  extern __shared__ float smem[];
  float* sW1 = smem;                    // 64 * 66
  float* sQ = sW1 + 64 * W1_PITCH;      // 224 * 66
  float* sQb = sQ + 224 * Q_PITCH;      // 224
  float* sPb0 = sQb + 224;              // 128
  float* sHbase = sPb0 + 128;           // 8 waves * 16*68
  float* sAttrBase = sHbase + 8 * 16 * H_PITCH;  // 8 waves * 16*9
  int* sDstBase = (int*)(sAttrBase + 8 * 16 * 9);  // 8 waves * 16

  const int tid = threadIdx.x;
  const int w = tid >> 5;
  const int lane = tid & 31;
  float* sH = sHbase + w * (16 * H_PITCH);
  float* sAttr = sAttrBase + w * (16 * 9);
  int* sDst = sDstBase + w * 16;

  // ---- stage folded weights into LDS (padded pitch, conflict-free b64) ----
  for (int idx = tid; idx < 64 * 64; idx += blockDim.x)
    sW1[(idx >> 6) * W1_PITCH + (idx & 63)] = W1t[idx];
  for (int idx = tid; idx < 224 * 64; idx += blockDim.x)
    sQ[(idx >> 6) * Q_PITCH + (idx & 63)] = Qt[idx];
  if (tid < 224) sQb[tid] = qb[tid];
  if (tid < 128) sPb0[tid] = proj_b0[tid];
  __syncthreads();

  const int hi = lane >> 4;   // half-wave select (K pair / row group)
  const int lm = lane & 15;   // row (A) / column (B,C,D)
  const int kb = 2 * hi;      // K base within a K-step

  // per-lane loop-invariant scalars (column = 16t + lm)
  float b1t[4], gt[4], bet[4];
#pragma unroll
  for (int t = 0; t < 4; ++t) {
    const int col = 16 * t + lm;
    b1t[t] = rad_b1[col];
    gt[t] = gamma[col];
    bet[t] = beta[col];
  }

  const int n_waves = gridDim.x * (blockDim.x >> 5);
  const int gw = blockIdx.x * (blockDim.x >> 5) + w;
  const int n_tiles = n_edges >> 4;

  for (int tile = gw; tile < n_tiles; tile += n_waves) {
    const int e0 = tile * 16;

    // ---- stage X tile (16 x 64) into sH, and edge attrs/dst ----
#pragma unroll
    for (int it = 0; it < 8; ++it) {
      const int idx = it * 32 + lane;
      const int m = idx >> 4;
      const int c4 = (idx & 15) << 2;
      const float4 v =
          *(const float4*)(edge_scalars + (size_t)(e0 + m) * 64 + c4);
      float* p = sH + m * H_PITCH + c4;
      p[0] = v.x;
      p[1] = v.y;
      p[2] = v.z;
      p[3] = v.w;
    }
    if (lane < 16) {
      sDst[lane] = edge_dst[e0 + lane] * 480;
      const float* ea = edge_attr + (size_t)(e0 + lane) * 9;
#pragma unroll
      for (int j = 0; j < 9; ++j) sAttr[lane * 9 + j] = ea[j];
    }
    // same-wave LDS ops are processed in order: no barrier needed.

    // ---- GEMM1: h = X @ W1 ----
    v8f acc[4];
#pragma unroll
    for (int t = 0; t < 4; ++t) {
      v8f c = {0.f, 0.f, 0.f, 0.f, 0.f, 0.f, 0.f, 0.f};
#pragma unroll
      for (int kk = 0; kk < 16; ++kk) {
        v2f a = *(const v2f*)(sH + lm * H_PITCH + 4 * kk + kb);
        v2f b = *(const v2f*)(sW1 + (16 * t + lm) * W1_PITCH + 4 * kk + kb);
        c = wmma4(a, b, c);
      }
      acc[t] = c;
    }

    // ---- LayerNorm(64) + affine + SiLU; write h back to sH (transpose) ----
#pragma unroll
    for (int i = 0; i < 8; ++i) {
      float h[4];
#pragma unroll
      for (int t = 0; t < 4; ++t) h[t] = acc[t][i] + b1t[t];
      float s1 = h[0] + h[1] + h[2] + h[3];
      float s2 = fmaf(h[0], h[0],
                      fmaf(h[1], h[1], fmaf(h[2], h[2], h[3] * h[3])));
#pragma unroll
      for (int m = 1; m < 16; m <<= 1) {
        s1 += __shfl_xor(s1, m, 32);
        s2 += __shfl_xor(s2, m, 32);
      }
      const float mu = s1 * (1.f / 64.f);
      const float var = s2 * (1.f / 64.f) - mu * mu;
      const float r = rsqrtf(var + 1e-5f);
      const int row = (i + 8 * hi) * H_PITCH + lm;
#pragma unroll
      for (int t = 0; t < 4; ++t) {
        float hn = (h[t] - mu) * r * gt[t] + bet[t];
        sH[row + 16 * t] = hn / (1.f + __expf(-hn));  // SiLU
      }
    }

    // ---- GEMM2: u = h @ Q (+ folded bias), fused scatter ----
#pragma unroll
    for (int t = 0; t < 14; ++t) {
      const float q = sQb[16 * t + lm];
      v8f c;
#pragma unroll
      for (int i = 0; i < 8; ++i) c[i] = q;
#pragma unroll
      for (int kk = 0; kk < 16; ++kk) {
        v2f a = *(const v2f*)(sH + lm * H_PITCH + 4 * kk + kb);
        v2f b = *(const v2f*)(sQ + (16 * t + lm) * Q_PITCH + 4 * kk + kb);
        c = wmma4(a, b, c);
      }
      // lane holds D(m = i + 8*hi, col = lm) for this N-tile t
      if (t < 8) {
        // y0[:, 16t+lm] = sh0*u0 + proj_b0 ; scaled by 0.25
        const float pb = sPb0[16 * t + lm];
#pragma unroll
        for (int i = 0; i < 8; ++i) {
          const int m = i + 8 * hi;
          const float val = sAttr[m * 9 + 0] * c[i] + pb;
          atomicAdd(out + sDst[m] + 16 * t + lm, 0.25f * val);
        }
      } else if (t < 12) {
        // y1[d*3+m] = u1[d]*sh1[m], d = 16*(t-8)+lm
        const int base = 128 + (16 * (t - 8) + lm) * 3;
#pragma unroll
        for (int i = 0; i < 8; ++i) {
          const int m = i + 8 * hi;
          const float u = 0.25f * c[i];
#pragma unroll
          for (int mm = 0; mm < 3; ++mm)
            atomicAdd(out + sDst[m] + base + mm, u * sAttr[m * 9 + 1 + mm]);
        }
      } else {
        // y2[d*5+m] = u2[d]*sh2[m], d = 16*(t-12)+lm
        const int base = 320 + (16 * (t - 12) + lm) * 5;
#pragma unroll
        for (int i = 0; i < 8; ++i) {
          const int m = i + 8 * hi;
          const float u = 0.25f * c[i];
#pragma unroll
          for (int mm = 0; mm < 5; ++mm)
            atomicAdd(out + sDst[m] + base + mm, u * sAttr[m * 9 + 4 + mm]);
        }
      }
    }
  }
}

// ---------------- host side ------------------------------------------------
extern "C" void kernel_launch(void* const* d_in, const int* in_sizes, int n_in,
                              void* d_out, int out_size, void* d_ws,
                              size_t ws_size, hipStream_t stream) {
  // input order: 0 node_input, 1 edge_attr, 2 edge_scalars, 3 edge_src,
  // 4 edge_dst, 5 batch, 6 exp_w, 7 exp_b, 8 rad_w1, 9 rad_b1, 10 rad_gamma,
  // 11 rad_beta, 12 rad_w2, 13 rad_b2, 14 proj_w0, 15 proj_b0, 16 proj_w1,
  // 17 proj_w2
  const float* edge_attr = (const float*)d_in[1];
  const float* edge_scalars = (const float*)d_in[2];
  const int* edge_dst = (const int*)d_in[4];
  const float* exp_w = (const float*)d_in[6];
  const float* exp_b = (const float*)d_in[7];
  const float* rad_w1 = (const float*)d_in[8];
  const float* rad_b1 = (const float*)d_in[9];
  const float* rad_gamma = (const float*)d_in[10];
  const float* rad_beta = (const float*)d_in[11];
  const float* rad_w2 = (const float*)d_in[12];
  const float* rad_b2 = (const float*)d_in[13];
  const float* proj_w0 = (const float*)d_in[14];
  const float* proj_b0 = (const float*)d_in[15];
  const float* proj_w1 = (const float*)d_in[16];
  const float* proj_w2 = (const float*)d_in[17];
  const int n_edges = in_sizes[4];

  float* ws = (float*)d_ws;
  float* Qt = ws;               // 224*64
  float* qb = Qt + 224 * 64;    // 224
  float* W1t = qb + 224;        // 64*64

  float* out = (float*)d_out;

  // zero output (atomically accumulated into)
  const int n4 = out_size >> 2;
  zero_out_kernel<<<(n4 + 255) / 256, 256, 0, stream>>>((float4*)out, n4);

  // fold weights
  fold_weights_kernel<<<292, 64, 0, stream>>>(exp_w, exp_b, rad_w1, rad_w2,
                                              rad_b2, proj_w0, proj_w1,
                                              proj_w2, Qt, qb, W1t);

  // fused edge network + scatter
  const size_t lds_bytes =
      (size_t)(64 * W1_PITCH + 224 * Q_PITCH + 224 + 128 + 8 * 16 * H_PITCH +
               8 * 16 * 9 + 8 * 16) *
      sizeof(float);
  edge_kernel<<<512, 256, lds_bytes, stream>>>(
      edge_attr, edge_scalars, edge_dst, rad_b1, rad_gamma, rad_beta, proj_b0,
      Qt, qb, W1t, out, n_edges);
}